// KuramotoHyperUniversal_37512244363645
// MI455X (gfx1250) — compile-verified
//
#include <hip/hip_runtime.h>
#include <cmath>
#include <cstdint>

// ---------------------------------------------------------------------------
// Kuramoto + MLP pipeline for MI455X (gfx1250), fp32 WMMA (16x16x4) GEMMs.
//
// Math rewrite: forcesum[b,i] = sum_j sin(yd[b,j]-yd[b,i]) * A[i,j]
//   = cos(yd[b,i]) * (sin(yd) @ A^T)[b,i] - sin(yd[b,i]) * (cos(yd) @ A^T)[b,i]
// so the O(dim^3) trig tensor becomes two 512^3 GEMMs.
//
// GEMM: 128x64 block tile, 8 waves, wave = 16x64 via 4 f32 WMMA accumulators.
// K-tiles are streamed into ping-pong LDS buffers with CDNA5 async
// global->LDS loads (ASYNCcnt), one barrier per tile. LDS tiles are
// k-contiguous with stride 20 so every fragment is one aligned ds_load_b64
// into an even VGPR pair (conflict-free banks, no repack moves).
// ---------------------------------------------------------------------------

typedef float v2f __attribute__((ext_vector_type(2)));
typedef float v8f __attribute__((ext_vector_type(8)));

#define DIMK   512
#define BATCH  512
#define HID    1538
#define INSZ   1537
#define LDPAD  1600   // padded leading dim for activation buffers (25*64)
#define KP     20     // LDS k-stride (pad): bank-conflict-free, 16B aligned rows

#define EPI_NONE 0
#define EPI_BIAS 1
#define EPI_TANH 2

// ---- CDNA5 async global->LDS helpers ---------------------------------------
__device__ __forceinline__ uint32_t lds_off(const void* p) {
    // AS(3) pointers: low 32 bits are the LDS byte offset (ISA 10.2).
    return (uint32_t)(size_t)p;
}
__device__ __forceinline__ void async_ld_b128(uint32_t lds, const float* g) {
    asm volatile("global_load_async_to_lds_b128 %0, %1, off"
                 :: "v"(lds), "v"((uint64_t)(size_t)g) : "memory");
}
__device__ __forceinline__ void async_ld_b32(uint32_t lds, const float* g) {
    asm volatile("global_load_async_to_lds_b32 %0, %1, off"
                 :: "v"(lds), "v"((uint64_t)(size_t)g) : "memory");
}
__device__ __forceinline__ void wait_async0() {
#if __has_builtin(__builtin_amdgcn_s_wait_asynccnt)
    __builtin_amdgcn_s_wait_asynccnt(0);
#else
    asm volatile("s_wait_asynccnt 0x0" ::: "memory");
#endif
}

// ---------------------------------------------------------------------------
// prep: mlp_in = [cos(yd) | sin(yd) | t-1 | freqs | 0-pad], plus sinY/cosY.
// ---------------------------------------------------------------------------
__global__ __launch_bounds__(256) void prep_kernel(
    const float* __restrict__ t, const float* __restrict__ y,
    const float* __restrict__ freqs,
    float* __restrict__ mlp_in, float* __restrict__ sinY, float* __restrict__ cosY)
{
    const int b = blockIdx.x;
    const float te = t[0] - 1.0f;
    float* row = mlp_in + (size_t)b * LDPAD;
    for (int j = threadIdx.x; j < DIMK; j += 256) {
        float yd = y[(size_t)b * (DIMK + 1) + j];
        float s, c;
        sincosf(yd, &s, &c);
        row[j]        = c;
        row[DIMK + j] = s;
        cosY[(size_t)b * DIMK + j] = c;
        sinY[(size_t)b * DIMK + j] = s;
        row[2 * DIMK + 1 + j] = freqs[(size_t)b * DIMK + j];
    }
    if (threadIdx.x == 0) row[2 * DIMK] = te;           // t column at 1024
    for (int j = INSZ + threadIdx.x; j < LDPAD; j += 256)
        row[j] = 0.0f;                                  // zero K-padding
}

// ---------------------------------------------------------------------------
// WMMA fp32 GEMM:  Y = epi( X[BATCH x K] * opB(W) + bias )
//   TRANSB=false: B = W (K x N, row-major, ldw)     [MLP layers]
//   TRANSB=true : B = W^T with W (N x K, ldw)       [sin/cos @ A^T]
// ---------------------------------------------------------------------------
template <bool TRANSB, int EPI>
__global__ __launch_bounds__(256) void gemm_wmma_kernel(
    const float* __restrict__ X, int ldx,
    const float* __restrict__ W, int ldw,
    const float* __restrict__ bias,
    float* __restrict__ Y, int ldy,
    int Kact, int Nact, int Nstore)
{
    __shared__ float sX[2][128][KP];  // ping-pong, [m][k] k-contiguous
    __shared__ float sW[2][64][KP];   // ping-pong, [n][k] k-contiguous

    const int tid  = threadIdx.x;
    const int lane = tid & 31;
    const int wv   = tid >> 5;
    const int hi   = lane >> 4;    // half-wave: K pair select / row+8
    const int l16  = lane & 15;

    const int mblk = blockIdx.y * 128;
    const int n0   = blockIdx.x * 64;

    const v8f vzero = {0.f, 0.f, 0.f, 0.f, 0.f, 0.f, 0.f, 0.f};
    v8f acc[4];
#pragma unroll
    for (int i = 0; i < 4; ++i) acc[i] = vzero;

    // ---- staging index maps (per thread)
    const int xr  = tid >> 2;         // X rows: xr and xr+64
    const int xc4 = (tid & 3) * 4;    // X k-quad: 0,4,8,12
    const int wn  = tid & 63;         // W (no-trans): n-local
    const int wk4 = (tid >> 6) * 4;   // W (no-trans): k-quad base
    const int tr  = tid >> 2;         // W (trans): n-local 0..63
    const int tc4 = (tid & 3) * 4;    // W (trans): k-quad

    // Clamped async tile fetch: OOB indices clamp to the last valid row/col.
    // Clamped K-tail rows of sW meet zero-padded X columns (finite*0 = 0);
    // clamped N-tail columns only feed output columns gn >= Nact, stored as 0.
    auto issue = [&](int kbase, int buf) {
        async_ld_b128(lds_off(&sX[buf][xr][xc4]),
                      X + (size_t)(mblk + xr) * ldx + kbase + xc4);
        async_ld_b128(lds_off(&sX[buf][xr + 64][xc4]),
                      X + (size_t)(mblk + 64 + xr) * ldx + kbase + xc4);
        if constexpr (!TRANSB) {
            const int ncol = min(n0 + wn, Nact - 1);
#pragma unroll
            for (int i = 0; i < 4; ++i) {
                const int krow = min(kbase + wk4 + i, Kact - 1);
                async_ld_b32(lds_off(&sW[buf][wn][wk4 + i]),
                             W + (size_t)krow * ldw + ncol);
            }
        } else {
            async_ld_b128(lds_off(&sW[buf][tr][tc4]),
                          W + (size_t)(n0 + tr) * ldw + kbase + tc4);
        }
    };

    const int KT = (Kact + 15) / 16;
    issue(0, 0);
    int cur = 0;
    for (int kt = 0; kt < KT; ++kt) {
        wait_async0();          // this wave's deposits done
        __syncthreads();        // all waves' deposits visible
        if (kt + 1 < KT) {
            const int knext = (kt + 1) * 16;
            issue(knext, cur ^ 1);             // overlaps with WMMA below
            if (kt + 2 < KT) {                 // prime L2 for the weight stream
                if constexpr (!TRANSB) {
                    __builtin_prefetch(W + (size_t)min(knext + 16 + wk4, Kact - 1) * ldw
                                         + min(n0 + wn, Nact - 1), 0, 3);
                } else {
                    __builtin_prefetch(W + (size_t)(n0 + tr) * ldw + knext + 16 + tc4, 0, 3);
                }
            }
        }

        // ---- 4 WMMA k-steps of 4, 4 n-tiles each
        const int mw = wv * 16;
#pragma unroll
        for (int kk = 0; kk < 16; kk += 4) {
            // A frag: lanes0-15 hold K=kk,kk+1; lanes16-31 hold K=kk+2,kk+3
            v2f a = *(const v2f*)&sX[cur][mw + l16][kk + hi * 2];
#pragma unroll
            for (int nt = 0; nt < 4; ++nt) {
                v2f b = *(const v2f*)&sW[cur][nt * 16 + l16][kk + hi * 2];
                acc[nt] = __builtin_amdgcn_wmma_f32_16x16x4_f32(
                    false, a, false, b, (short)0, acc[nt], false, false);
            }
        }
        cur ^= 1;
    }

    // ---- epilogue: C/D layout: VGPR e -> row M = e + hi*8, col N = l16
    const int gm0 = mblk + wv * 16 + hi * 8;
#pragma unroll
    for (int nt = 0; nt < 4; ++nt) {
        const int gn = n0 + nt * 16 + l16;
        if (gn >= Nstore) continue;
        float bv = 0.f;
        if constexpr (EPI != EPI_NONE) {
            if (gn < Nact) bv = bias[gn];
        }
#pragma unroll
        for (int e = 0; e < 8; ++e) {
            float v = acc[nt][e];
            if constexpr (EPI == EPI_TANH) {
                v = (gn < Nact) ? tanhf(v + bv) : 0.0f;   // zero the N padding
            } else if constexpr (EPI == EPI_BIAS) {
                v = v + bv;
            }
            Y[(size_t)(gm0 + e) * ldy + gn] = v;
        }
    }
}

// ---------------------------------------------------------------------------
// finalize: force = cforce * (cosY*S - sinY*C) / dim + freqs ; f1 = sum cf^2
// out row b = [force[0..511], f1], row length 513.
// ---------------------------------------------------------------------------
__global__ __launch_bounds__(256) void finalize_kernel(
    const float* __restrict__ cforce, const float* __restrict__ sinY,
    const float* __restrict__ cosY, const float* __restrict__ S,
    const float* __restrict__ C, const float* __restrict__ freqs,
    float* __restrict__ out)
{
    __shared__ float red[8];
    const int b = blockIdx.x;
    const size_t rb = (size_t)b * DIMK;
    float ss = 0.f;
    for (int i = threadIdx.x; i < DIMK; i += 256) {
        const float cf = cforce[rb + i];
        const float fs = cosY[rb + i] * S[rb + i] - sinY[rb + i] * C[rb + i];
        out[(size_t)b * (DIMK + 1) + i] = cf * fs * (1.0f / DIMK) + freqs[rb + i];
        ss += cf * cf;
    }
#pragma unroll
    for (int off = 16; off > 0; off >>= 1) ss += __shfl_down(ss, off, 32);
    if ((threadIdx.x & 31) == 0) red[threadIdx.x >> 5] = ss;
    __syncthreads();
    if (threadIdx.x == 0) {
        float v = 0.f;
#pragma unroll
        for (int i = 0; i < 8; ++i) v += red[i];
        out[(size_t)b * (DIMK + 1) + DIMK] = v;
    }
}

// ---------------------------------------------------------------------------
extern "C" void kernel_launch(void* const* d_in, const int* in_sizes, int n_in,
                              void* d_out, int out_size, void* d_ws, size_t ws_size,
                              hipStream_t stream)
{
    const float* t  = (const float*)d_in[0];
    const float* y  = (const float*)d_in[1];
    const float* fr = (const float*)d_in[2];
    const float* A  = (const float*)d_in[3];
    const float* W0 = (const float*)d_in[4];
    const float* b0 = (const float*)d_in[5];
    const float* W1 = (const float*)d_in[6];
    const float* b1 = (const float*)d_in[7];
    const float* W2 = (const float*)d_in[8];
    const float* b2 = (const float*)d_in[9];
    const float* W3 = (const float*)d_in[10];
    const float* b3 = (const float*)d_in[11];
    float* out = (float*)d_out;

    float* ws      = (float*)d_ws;
    float* mlp_in  = ws;  ws += (size_t)BATCH * LDPAD;
    float* hA      = ws;  ws += (size_t)BATCH * LDPAD;
    float* hB      = ws;  ws += (size_t)BATCH * LDPAD;
    float* cforce  = ws;  ws += (size_t)BATCH * DIMK;
    float* sinY    = ws;  ws += (size_t)BATCH * DIMK;
    float* cosY    = ws;  ws += (size_t)BATCH * DIMK;
    float* Sb      = ws;  ws += (size_t)BATCH * DIMK;
    float* Cb      = ws;  ws += (size_t)BATCH * DIMK;

    prep_kernel<<<BATCH, 256, 0, stream>>>(t, y, fr, mlp_in, sinY, cosY);

    const dim3 blk(256);
    const dim3 gHid((HID + 63) / 64, BATCH / 128);   // 25 x 4
    const dim3 gDim((DIMK + 63) / 64, BATCH / 128);  //  8 x 4

    // h0 = tanh(mlp_in @ W0 + b0)
    gemm_wmma_kernel<false, EPI_TANH><<<gHid, blk, 0, stream>>>(
        mlp_in, LDPAD, W0, HID, b0, hA, LDPAD, INSZ, HID, LDPAD);
    // h1 = tanh(h0 @ W1 + b1)
    gemm_wmma_kernel<false, EPI_TANH><<<gHid, blk, 0, stream>>>(
        hA, LDPAD, W1, HID, b1, hB, LDPAD, HID, HID, LDPAD);
    // h2 = tanh(h1 @ W2 + b2)
    gemm_wmma_kernel<false, EPI_TANH><<<gHid, blk, 0, stream>>>(
        hB, LDPAD, W2, HID, b2, hA, LDPAD, HID, HID, LDPAD);
    // cforce = h2 @ W3 + b3
    gemm_wmma_kernel<false, EPI_BIAS><<<gDim, blk, 0, stream>>>(
        hA, LDPAD, W3, DIMK, b3, cforce, DIMK, HID, DIMK, DIMK);
    // S = sinY @ A^T ; C = cosY @ A^T
    gemm_wmma_kernel<true, EPI_NONE><<<gDim, blk, 0, stream>>>(
        sinY, DIMK, A, DIMK, nullptr, Sb, DIMK, DIMK, DIMK, DIMK);
    gemm_wmma_kernel<true, EPI_NONE><<<gDim, blk, 0, stream>>>(
        cosY, DIMK, A, DIMK, nullptr, Cb, DIMK, DIMK, DIMK, DIMK);

    finalize_kernel<<<BATCH, blk, 0, stream>>>(cforce, sinY, cosY, Sb, Cb, fr, out);
}